// BayesianNet_29935922053298
// MI455X (gfx1250) — compile-verified
//
#include <hip/hip_runtime.h>
#include <math.h>
#include <stdint.h>

// ---------------------------------------------------------------------------
// Bayesian MLP forward, fused weight-sampling + fp32 WMMA GEMM for gfx1250.
//   M = 64 (batch), layers: 1024->4096, 4x (4096->4096), 4096->1024
//   w = mu + N(0,1)*softplus(rho) + 1e-6 ; KL accumulated over all params.
// Roofline: ~604 MB of (mu,rho) traffic dominates (~26us @ 23.3 TB/s);
// GEMM is only 9.7 GFLOP fp32. Single streaming pass over mu/rho with
// non-temporal loads; sampled weights live only in registers as WMMA
// B-operands; A staged through LDS (conflict-free padded layout, read back
// as ds_load_2addr_b64); all transcendentals use raw hardware ops
// (v_exp/v_log/v_cos/v_sqrt/v_rcp) to avoid libm guard chains; KL reduced
// deterministically (no float atomics -> bitwise replay-stable).
// ---------------------------------------------------------------------------

typedef __attribute__((ext_vector_type(2))) float v2f;
typedef __attribute__((ext_vector_type(8))) float v8f;

#define BAYES_EPS 1e-6f
#define KCHUNK 64
#define LDS_STRIDE 68   // 64 + 4 pad: 272B rows -> 16B aligned, bank-conflict-free

// ---- raw hardware transcendentals (single instruction each) ----
#if __has_builtin(__builtin_amdgcn_logf)
__device__ __forceinline__ float fast_log(float x) {
  return 0.69314718056f * __builtin_amdgcn_logf(x);   // v_log_f32 (log2)
}
#else
__device__ __forceinline__ float fast_log(float x) { return __logf(x); }
#endif

#if __has_builtin(__builtin_amdgcn_exp2f)
__device__ __forceinline__ float fast_exp(float x) {
  return __builtin_amdgcn_exp2f(1.44269504089f * x);  // v_exp_f32 (exp2)
}
#else
__device__ __forceinline__ float fast_exp(float x) { return __expf(x); }
#endif

#if __has_builtin(__builtin_amdgcn_rcpf)
__device__ __forceinline__ float fast_rcp(float x) {
  return __builtin_amdgcn_rcpf(x);                    // v_rcp_f32
}
#else
__device__ __forceinline__ float fast_rcp(float x) { return 1.0f / x; }
#endif

#if __has_builtin(__builtin_amdgcn_sqrtf)
__device__ __forceinline__ float fast_sqrt(float x) {
  return __builtin_amdgcn_sqrtf(x);                   // v_sqrt_f32
}
#else
__device__ __forceinline__ float fast_sqrt(float x) { return __fsqrt_rn(x); }
#endif

// v_cos_f32 input is in revolutions: returns cos(2*pi*x). No range reduction.
#if __has_builtin(__builtin_amdgcn_cosf)
__device__ __forceinline__ float cos_2pi(float x) {
  return __builtin_amdgcn_cosf(x);
}
#else
__device__ __forceinline__ float cos_2pi(float x) {
  return __cosf(6.28318530718f * x);
}
#endif

// lowbias32 hash: 2x 32-bit mul + shifts, dual-issue friendly.
__device__ __forceinline__ uint32_t hash32(uint32_t x) {
  x ^= x >> 16;
  x *= 0x7FEB352Du;
  x ^= x >> 15;
  x *= 0x846CA68Bu;
  x ^= x >> 16;
  return x;
}

// Deterministic counter-based standard normal (Box-Muller, cosine branch).
__device__ __forceinline__ float gauss(uint32_t seed, uint32_t idx) {
  const uint32_t u = hash32(idx ^ (seed * 0x9E3779B9u));
  const uint32_t v = hash32(u + 0x9E3779B9u);
  const float u1 = (float)(u >> 8) * 0x1p-24f + 0x1p-25f;  // (0,1)
  const float u2 = (float)(v >> 8) * 0x1p-24f;             // [0,1)
  const float r = fast_sqrt(-2.0f * fast_log(u1));
  return r * cos_2pi(u2);
}

// Stable, branch-free softplus: log1p(e^x) = max(x,0) + log(1+e^-|x|).
__device__ __forceinline__ float softplusf(float x) {
  return fmaxf(x, 0.0f) + fast_log(1.0f + fast_exp(-fabsf(x)));
}

// KL contribution given sampled gaussian g, weight w, sigma:
//   d = (w - mu)/sigma = g + eps/sigma ;  kl = -0.5 d^2 - log(sigma) + 0.5 w^2
__device__ __forceinline__ float kl_term(float g, float w, float sg) {
  const float d = fmaf(BAYES_EPS, fast_rcp(sg), g);
  return fmaf(-0.5f, d * d, fmaf(0.5f, w * w, -fast_log(sg)));
}

// Deterministic block KL reduction: wave shuffle -> LDS -> thread 0 sums.
__device__ __forceinline__ void block_kl_reduce(float kl, float* partial_slot) {
  __shared__ float red[8];
  #pragma unroll
  for (int off = 16; off > 0; off >>= 1) kl += __shfl_xor(kl, off, 32);
  const int lane = threadIdx.x & 31;
  const int wv = threadIdx.x >> 5;
  if (lane == 0) red[wv] = kl;
  __syncthreads();
  if (threadIdx.x == 0) {
    float s = 0.0f;
    const int nw = (blockDim.x + 31) >> 5;
    for (int i = 0; i < nw; ++i) s += red[i];
    *partial_slot = s;
  }
}

// ---------------------------------------------------------------------------
// Bias sampling: b = mu + N*softplus(rho) + eps, plus its KL partial.
// ---------------------------------------------------------------------------
__global__ __launch_bounds__(256) void bayes_bias_kernel(
    const float* __restrict__ mu, const float* __restrict__ rho,
    float* __restrict__ b_out, float* __restrict__ partial_slot,
    int n, uint32_t seed) {
  float kl = 0.0f;
  for (int i = threadIdx.x; i < n; i += 256) {
    float m = mu[i];
    float sg = softplusf(rho[i]);
    float g = gauss(seed, (uint32_t)i);
    float w = fmaf(g, sg, m) + BAYES_EPS;
    b_out[i] = w;
    kl += kl_term(g, w, sg);
  }
  block_kl_reduce(kl, partial_slot);
}

// ---------------------------------------------------------------------------
// Fused sample + GEMM + bias (+ReLU) using V_WMMA_F32_16X16X4_F32.
//   out[64 x N] = act(A[64 x K] @ sample(Wmu,Wrho)[K x N] + b)
// Block: 128 threads = 4 waves; each wave owns a 64x16 output strip
// (4 accumulators); block covers N=64; grid.x = N/64.
// A is staged in 64x64 chunks into LDS (cooperative, coalesced b128),
// fragments read back as conflict-free ds_load_2addr_b64. Weight mu/rho are
// a single-use non-temporal stream; each (k,n) sampled/KL-counted once.
//
// Operand layouts (ISA 7.12.2, wave32):
//   A 16x4 f32 : lanes 0-15 M=0..15 {VGPR0:K=0, VGPR1:K=1}; lanes 16-31 {K=2,K=3}
//   B 4x16 f32 : lanes 0-15 N=0..15 {VGPR0:K=0, VGPR1:K=1}; lanes 16-31 {K=2,K=3}
//   C/D 16x16  : VGPR r -> M=r (lanes 0-15) / M=r+8 (lanes 16-31), N=lane%16
// ---------------------------------------------------------------------------
__global__ __launch_bounds__(128) void bayes_gemm_kernel(
    const float* __restrict__ A, const float* __restrict__ wmu,
    const float* __restrict__ wrho, const float* __restrict__ bias,
    float* __restrict__ out, float* __restrict__ partial_base,
    int K, int N, uint32_t seed, int relu) {
  __shared__ float lds_a[64 * LDS_STRIDE];

  const int lane   = threadIdx.x & 31;
  const int wave   = threadIdx.x >> 5;
  const int n_base = blockIdx.x * 64 + wave * 16;
  const int m      = lane & 15;        // M row within tile / N col for B
  const int khalf  = (lane >> 4) * 2;  // 0 or 2

  uint32_t wi = (uint32_t)(khalf * N + n_base + m);
  const uint32_t wstep = 4u * (uint32_t)N;

  v8f acc0 = {0.f, 0.f, 0.f, 0.f, 0.f, 0.f, 0.f, 0.f};
  v8f acc1 = acc0, acc2 = acc0, acc3 = acc0;
  float kl = 0.0f;

  const float4* __restrict__ A4 = (const float4*)A;
  const uint32_t K4 = (uint32_t)K >> 2;

  for (int kc = 0; kc < K; kc += KCHUNK) {
    // ---- cooperative stage: A[0:64, kc:kc+64] -> LDS (b128 in/out) ----
    #pragma unroll
    for (int i = 0; i < 8; ++i) {
      const int j  = (int)threadIdx.x + i * 128;  // 0..1023 float4 slots
      const int r  = j >> 4;                      // row 0..63
      const int c4 = j & 15;                      // col4 0..15
      const float4 v = A4[(uint32_t)r * K4 + (uint32_t)(kc >> 2) + (uint32_t)c4];
      *(float4*)&lds_a[r * LDS_STRIDE + c4 * 4] = v;
    }
    __syncthreads();

    #pragma unroll 2
    for (int kk = 0; kk < KCHUNK; kk += 4) {
      const int kl_ofs = kk + khalf;  // K offset within chunk for this lane

      // ---- A fragments from LDS (ds_load_2addr_b64, conflict-free) ----
      const v2f a0 = *(const v2f*)&lds_a[(0 * 16 + m) * LDS_STRIDE + kl_ofs];
      const v2f a1 = *(const v2f*)&lds_a[(1 * 16 + m) * LDS_STRIDE + kl_ofs];
      const v2f a2 = *(const v2f*)&lds_a[(2 * 16 + m) * LDS_STRIDE + kl_ofs];
      const v2f a3 = *(const v2f*)&lds_a[(3 * 16 + m) * LDS_STRIDE + kl_ofs];

      // ---- weight mu/rho: single-use stream -> non-temporal loads ----
      const float mu0 = __builtin_nontemporal_load(wmu + wi);
      const float rh0 = __builtin_nontemporal_load(wrho + wi);
      const float mu1 = __builtin_nontemporal_load(wmu + wi + (uint32_t)N);
      const float rh1 = __builtin_nontemporal_load(wrho + wi + (uint32_t)N);

      // ---- sample 4x16 weight tile (2 elements per lane) + KL ----
      const float sg0 = softplusf(rh0);
      const float sg1 = softplusf(rh1);
      const float g0 = gauss(seed, wi);
      const float g1 = gauss(seed, wi + (uint32_t)N);
      const float w0 = fmaf(g0, sg0, mu0) + BAYES_EPS;
      const float w1 = fmaf(g1, sg1, mu1) + BAYES_EPS;
      kl += kl_term(g0, w0, sg0) + kl_term(g1, w1, sg1);
      wi += wstep;

      v2f b;
      b.x = w0;
      b.y = w1;

      // ---- 4 WMMAs sharing the sampled B tile ----
      acc0 = __builtin_amdgcn_wmma_f32_16x16x4_f32(false, a0, false, b,
                                                   (short)0, acc0, false, false);
      acc1 = __builtin_amdgcn_wmma_f32_16x16x4_f32(false, a1, false, b,
                                                   (short)0, acc1, false, false);
      acc2 = __builtin_amdgcn_wmma_f32_16x16x4_f32(false, a2, false, b,
                                                   (short)0, acc2, false, false);
      acc3 = __builtin_amdgcn_wmma_f32_16x16x4_f32(false, a3, false, b,
                                                   (short)0, acc3, false, false);
    }
    __syncthreads();
  }

  // ---- epilogue: bias, ReLU, store ----
  const int col = lane & 15;
  const int rext = (lane >> 4) * 8;  // upper half-lanes hold M=r+8
  const float bv = bias[n_base + col];
  #pragma unroll
  for (int t = 0; t < 4; ++t) {
    const v8f accv = (t == 0) ? acc0 : (t == 1) ? acc1 : (t == 2) ? acc2 : acc3;
    #pragma unroll
    for (int r = 0; r < 8; ++r) {
      float v = accv[r] + bv;
      if (relu) v = fmaxf(v, 0.0f);
      const int row = t * 16 + rext + r;
      out[(size_t)row * N + n_base + col] = v;
    }
  }

  block_kl_reduce(kl, partial_base + blockIdx.x);
}

// Sequential deterministic final KL sum.
__global__ void kl_reduce_kernel(const float* __restrict__ partials, int n,
                                 float* __restrict__ kl_out) {
  if (threadIdx.x == 0 && blockIdx.x == 0) {
    float s = 0.0f;
    for (int i = 0; i < n; ++i) s += partials[i];
    *kl_out = s;
  }
}

// ---------------------------------------------------------------------------
extern "C" void kernel_launch(void* const* d_in, const int* in_sizes, int n_in,
                              void* d_out, int out_size, void* d_ws,
                              size_t ws_size, hipStream_t stream) {
  (void)in_sizes; (void)n_in; (void)out_size; (void)ws_size;
  constexpr int D_IN = 1024, D_H = 4096, D_OUT = 1024, N_HID = 4, M = 64;

  const float* x      = (const float*)d_in[0];
  const float* w1_mu  = (const float*)d_in[1];
  const float* w1_rho = (const float*)d_in[2];
  const float* b1_mu  = (const float*)d_in[3];
  const float* b1_rho = (const float*)d_in[4];
  const float* wh_mu  = (const float*)d_in[5];
  const float* wh_rho = (const float*)d_in[6];
  const float* bh_mu  = (const float*)d_in[7];
  const float* bh_rho = (const float*)d_in[8];
  const float* wf_mu  = (const float*)d_in[9];
  const float* wf_rho = (const float*)d_in[10];
  const float* bf_mu  = (const float*)d_in[11];
  const float* bf_rho = (const float*)d_in[12];

  float* y_out  = (float*)d_out;                 // 64 x 1024
  float* kl_out = (float*)d_out + (size_t)M * D_OUT;

  // Workspace: two activation buffers, one bias buffer, KL partial slots.
  float* ws    = (float*)d_ws;
  float* act0  = ws;                             // 64*4096
  float* act1  = ws + (size_t)M * D_H;           // 64*4096
  float* biasb = ws + 2 * (size_t)M * D_H;       // 4096
  float* parts = biasb + D_H;                    // partial KL slots

  int slot = 0;

  // Layer 0: 1024 -> 4096, ReLU
  bayes_bias_kernel<<<1, 256, 0, stream>>>(b1_mu, b1_rho, biasb, parts + slot,
                                           D_H, /*seed=*/1u);
  slot += 1;
  bayes_gemm_kernel<<<D_H / 64, 128, 0, stream>>>(
      x, w1_mu, w1_rho, biasb, act0, parts + slot, D_IN, D_H, /*seed=*/0u, 1);
  slot += D_H / 64;

  // Hidden layers: 4096 -> 4096, ReLU
  const float* cur = act0;
  float* nxt = act1;
  for (int i = 0; i < N_HID; ++i) {
    const size_t wofs = (size_t)i * D_H * D_H;
    const size_t bofs = (size_t)i * D_H;
    bayes_bias_kernel<<<1, 256, 0, stream>>>(bh_mu + bofs, bh_rho + bofs, biasb,
                                             parts + slot,
                                             D_H, /*seed=*/2u * (i + 1) + 1u);
    slot += 1;
    bayes_gemm_kernel<<<D_H / 64, 128, 0, stream>>>(
        cur, wh_mu + wofs, wh_rho + wofs, biasb, nxt, parts + slot, D_H, D_H,
        /*seed=*/2u * (i + 1), 1);
    slot += D_H / 64;
    const float* t = cur; cur = nxt; nxt = (float*)t;
  }

  // Final layer: 4096 -> 1024, no ReLU, writes y directly to d_out.
  bayes_bias_kernel<<<1, 256, 0, stream>>>(bf_mu, bf_rho, biasb, parts + slot,
                                           D_OUT, /*seed=*/11u);
  slot += 1;
  bayes_gemm_kernel<<<D_OUT / 64, 128, 0, stream>>>(
      cur, wf_mu, wf_rho, biasb, y_out, parts + slot, D_H, D_OUT,
      /*seed=*/10u, 0);
  slot += D_OUT / 64;

  kl_reduce_kernel<<<1, 32, 0, stream>>>(parts, slot, kl_out);
}